// GenerateGraph_21534966022861
// MI455X (gfx1250) — compile-verified
//
#include <hip/hip_runtime.h>

// Problem constants from setup_inputs(): B=16, N=4096, C=64, S=1024, K=32.
constexpr int B = 16;
constexpr int N = 4096;
constexpr int C = 64;
constexpr int S = 1024;   // fps_points
constexpr int K = 32;     // neighbors

#define INF_F 1e10f

typedef __attribute__((ext_vector_type(2))) float v2f;
typedef __attribute__((ext_vector_type(8))) float v8f;

// ---------------------------------------------------------------------------
// Kernel 1: farthest point sampling. One workgroup (1024 threads = 32 waves)
// per batch. xyz for the batch staged in LDS (48 KB dynamic, float4 copies),
// running min-distances in registers (4 points / thread).
//
// ONE __syncthreads per serial step: per-wave argmax partials go into a
// double-buffered LDS array; after the barrier every wave redundantly
// butterfly-reduces the 32 partials (32 lanes read 32 consecutive words ->
// conflict-free), so all waves agree on the winner with no second barrier.
// ---------------------------------------------------------------------------
__global__ __launch_bounds__(1024)
void fps_kernel(const float* __restrict__ xyz,
                const int* __restrict__ farthest0,
                int* __restrict__ fps_idx,
                float* __restrict__ new_xyz)
{
    const int b    = blockIdx.x;
    const int tid  = threadIdx.x;       // 0..1023
    const int lane = tid & 31;
    const int wave = tid >> 5;          // 0..31

    extern __shared__ float xyzS[];     // [N*3] = 48 KB
    __shared__ float redV[2][32];
    __shared__ int   redI[2][32];

    const float* xb = xyz + (size_t)b * N * 3;
    {   // 128-bit staging copies: N*3/4 = 3072 float4s, 3 per thread
        const float4* src = reinterpret_cast<const float4*>(xb);
        float4*       dst = reinterpret_cast<float4*>(xyzS);
        for (int i = tid; i < N * 3 / 4; i += 1024) dst[i] = src[i];
    }
    int far = farthest0[b];             // uniform broadcast load
    __syncthreads();

    // 4 points per thread, contiguous so global argmax tie-break = lowest idx.
    const int base = tid * 4;
    float px[4], py[4], pz[4], dist[4];
#pragma unroll
    for (int j = 0; j < 4; ++j) {
        px[j]   = xyzS[(base + j) * 3 + 0];
        py[j]   = xyzS[(base + j) * 3 + 1];
        pz[j]   = xyzS[(base + j) * 3 + 2];
        dist[j] = INF_F;
    }

    int p = 0;                          // parity for double-buffered partials
    for (int it = 0; it < S; ++it) {
        if (tid == 0) {
            // reference emits the *incoming* farthest each step
            fps_idx[b * S + it] = far;
            new_xyz[((size_t)b * S + it) * 3 + 0] = xyzS[far * 3 + 0];
            new_xyz[((size_t)b * S + it) * 3 + 1] = xyzS[far * 3 + 1];
            new_xyz[((size_t)b * S + it) * 3 + 2] = xyzS[far * 3 + 2];
        }
        const float cx = xyzS[far * 3 + 0];
        const float cy = xyzS[far * 3 + 1];
        const float cz = xyzS[far * 3 + 2];

        float bestV = -1.0f;
        int   bestI = base;
#pragma unroll
        for (int j = 0; j < 4; ++j) {
            const float dx = px[j] - cx, dy = py[j] - cy, dz = pz[j] - cz;
            const float d  = dx * dx + dy * dy + dz * dz;
            const float nd = fminf(dist[j], d);
            dist[j] = nd;
            if (nd > bestV) { bestV = nd; bestI = base + j; }
        }
        // wave32 butterfly argmax (prefer larger value, tie -> lower index)
#pragma unroll
        for (int m = 16; m >= 1; m >>= 1) {
            const float ov = __shfl_xor(bestV, m, 32);
            const int   oi = __shfl_xor(bestI, m, 32);
            if (ov > bestV || (ov == bestV && oi < bestI)) { bestV = ov; bestI = oi; }
        }
        if (lane == 0) { redV[p][wave] = bestV; redI[p][wave] = bestI; }
        __syncthreads();
        // every wave reduces the 32 partials itself -> all agree, no 2nd barrier
        float bv = redV[p][lane];
        int   bi = redI[p][lane];
#pragma unroll
        for (int m = 16; m >= 1; m >>= 1) {
            const float ov = __shfl_xor(bv, m, 32);
            const int   oi = __shfl_xor(bi, m, 32);
            if (ov > bv || (ov == bv && oi < bi)) { bv = ov; bi = oi; }
        }
        far = bi;
        p ^= 1;
    }
}

// ---------------------------------------------------------------------------
// Kernel 2: squared distance via V_WMMA_F32_16X16X4_F32 + iterative top-K.
// One workgroup (128 threads = 4 waves) per (batch, 16-row s-tile).
//
// WMMA mapping (K=3 padded to 4, norms folded in):
//   A[m][*] = (-2 sx, -2 sy, -2 sz, |s|^2)      (16x4, 2 VGPRs/lane)
//   B[*][n] = ( nx,    ny,    nz,   1    )      (4x16, 2 VGPRs/lane)
//   C[m][n] = |xyz_n|^2  (broadcast down column n)
//   D = A*B + C = |s|^2 + |n|^2 - 2 s.n  == squared distance, exactly.
//
// DSTRIDE = N+12: within each D-fragment ds_store (fixed VGPR i) the 32
// lanes hit 32 distinct banks, and in the phase-2 strided scan the
// (row,lane) -> bank map {12*row + cc (+8s)} is also conflict-free.
// ---------------------------------------------------------------------------
constexpr int DSTRIDE = N + 12;  // padded LDS row stride (floats)

__global__ __launch_bounds__(128)
void dist_topk_kernel(const float* __restrict__ xyz,
                      const float* __restrict__ new_xyz,
                      int* __restrict__ knn_idx)
{
    const int b    = blockIdx.x;
    const int st   = blockIdx.y;        // s-tile index, 16 rows
    const int tid  = threadIdx.x;       // 0..127
    const int lane = tid & 31;
    const int wave = tid >> 5;

    extern __shared__ float distS[];    // [16 * DSTRIDE]

    const float* xb = xyz + (size_t)b * N * 3;
    const float* sb = new_xyz + ((size_t)b * S + (size_t)st * 16) * 3;

    // ---- A fragment: shared across all n-tiles of this wave ----
    const int   mrow  = lane & 15;
    const float sx    = sb[mrow * 3 + 0];
    const float sy    = sb[mrow * 3 + 1];
    const float sz    = sb[mrow * 3 + 2];
    const float snorm = sx * sx + sy * sy + sz * sz;
    v2f a;
    a.x = (lane < 16) ? (-2.0f * sx) : (-2.0f * sz);
    a.y = (lane < 16) ? (-2.0f * sy) : snorm;

    // ---- sweep the 256 n-tiles; 64 per wave ----
    for (int t = wave; t < N / 16; t += 4) {
        const int   ncol = t * 16 + (lane & 15);
        const float nx   = xb[ncol * 3 + 0];
        const float ny   = xb[ncol * 3 + 1];
        const float nz   = xb[ncol * 3 + 2];
        v2f bm;
        bm.x = (lane < 16) ? nx : nz;
        bm.y = (lane < 16) ? ny : 1.0f;
        const float nnorm = nx * nx + ny * ny + nz * nz;
        v8f c;
#pragma unroll
        for (int i = 0; i < 8; ++i) c[i] = nnorm;

        v8f d = __builtin_amdgcn_wmma_f32_16x16x4_f32(
                    /*neg_a=*/false, a, /*neg_b=*/false, bm,
                    /*c_mod=*/(short)0, c, /*reuse_a=*/false, /*reuse_b=*/false);

        const int rbase = (lane < 16) ? 0 : 8;
#pragma unroll
        for (int i = 0; i < 8; ++i)
            distS[(rbase + i) * DSTRIDE + ncol] = d[i];
    }
    __syncthreads();

    // ---- top-K smallest per row: 16 rows x 8 lanes, K argmin rounds.
    //      4 independent sub-streams per lane break the cmp/select chain. ----
    const int row = tid >> 3;           // 0..15
    const int cc  = tid & 7;            // 0..7 (lane-interleaved columns)
    float* drow = distS + row * DSTRIDE;
    int*   orow = knn_idx + (((size_t)b * S) + (size_t)st * 16 + row) * K;

    for (int k = 0; k < K; ++k) {
        float bv4[4];
        int   bi4[4];
#pragma unroll
        for (int s = 0; s < 4; ++s) { bv4[s] = 2.0f * INF_F; bi4[s] = cc + 8 * s; }
        for (int n = cc; n < N; n += 32) {
#pragma unroll
            for (int s = 0; s < 4; ++s) {
                const int   ns = n + 8 * s;
                const float v  = drow[ns];
                if (v < bv4[s]) { bv4[s] = v; bi4[s] = ns; }
            }
        }
        float bv = bv4[0];
        int   bi = bi4[0];
#pragma unroll
        for (int s = 1; s < 4; ++s)
            if (bv4[s] < bv || (bv4[s] == bv && bi4[s] < bi)) { bv = bv4[s]; bi = bi4[s]; }
        // merge the 8 lanes of this row (aligned group -> xor stays in group)
#pragma unroll
        for (int m = 4; m >= 1; m >>= 1) {
            const float ov = __shfl_xor(bv, m, 32);
            const int   oi = __shfl_xor(bi, m, 32);
            if (ov < bv || (ov == bv && oi < bi)) { bv = ov; bi = oi; }
        }
        if ((bi & 7) == cc) drow[bi] = 2.0f * INF_F;   // owner poisons winner
        if (cc == 0) orow[k] = bi;
    }
}

// ---------------------------------------------------------------------------
// Kernel 3: edge = x[knn_idx] - x[fps_idx][None]. Pure-bandwidth gather,
// fully float4-vectorized (b128 loads/stores). One block per (b,s).
// ---------------------------------------------------------------------------
__global__ __launch_bounds__(256)
void edge_kernel(const float* __restrict__ x,
                 const int* __restrict__ fps_idx,
                 const int* __restrict__ knn_idx,
                 float* __restrict__ edge)
{
    const int bs  = blockIdx.x;         // b*S + s
    const int b   = bs >> 10;           // / S (S == 1024)
    const int tid = threadIdx.x;

    __shared__ float4 centerS[C / 4];   // 16 float4s
    __shared__ int    idxS[K];

    const float4* xb4 = reinterpret_cast<const float4*>(x + (size_t)b * N * C);
    const int     ctr = fps_idx[bs];
    if (tid < C / 4) centerS[tid] = xb4[(size_t)ctr * (C / 4) + tid];
    if (tid < K)     idxS[tid]    = knn_idx[(size_t)bs * K + tid];
    __syncthreads();

    float4* out4 = reinterpret_cast<float4*>(edge + (size_t)bs * K * C);
    const int nvec = K * C / 4;         // 512 float4s
    for (int e = tid; e < nvec; e += 256) {
        const int k  = e >> 4;          // / (C/4)
        const int c4 = e & 15;
        const float4 g = xb4[(size_t)idxS[k] * (C / 4) + c4];
        const float4 c = centerS[c4];
        float4 o;
        o.x = g.x - c.x; o.y = g.y - c.y; o.z = g.z - c.z; o.w = g.w - c.w;
        out4[e] = o;
    }
}

// ---------------------------------------------------------------------------
extern "C" void kernel_launch(void* const* d_in, const int* in_sizes, int n_in,
                              void* d_out, int out_size, void* d_ws, size_t ws_size,
                              hipStream_t stream)
{
    (void)in_sizes; (void)n_in; (void)out_size; (void)ws_size;

    const float* xyz  = (const float*)d_in[0];   // [B,N,3] f32
    const float* x    = (const float*)d_in[1];   // [B,N,C] f32
    const int*   far0 = (const int*)d_in[2];     // [B]     i32
    // d_in[3] = fps_points (1024), d_in[4] = neighbors (32): compile-time here.

    float* new_xyz = (float*)d_out;                          // B*S*3
    float* edge    = (float*)d_out + (size_t)B * S * 3;      // B*S*K*C

    int* fps_idx = (int*)d_ws;                               // B*S ints
    int* knn_idx = fps_idx + (size_t)B * S;                  // B*S*K ints

    fps_kernel<<<B, 1024, (size_t)N * 3 * sizeof(float), stream>>>(
        xyz, far0, fps_idx, new_xyz);

    dist_topk_kernel<<<dim3(B, S / 16), 128,
                       (size_t)16 * DSTRIDE * sizeof(float), stream>>>(
        xyz, new_xyz, knn_idx);

    edge_kernel<<<B * S, 256, 0, stream>>>(x, fps_idx, knn_idx, edge);
}